// GroundTruthBackward_21947282883151
// MI455X (gfx1250) — compile-verified
//
#include <hip/hip_runtime.h>
#include <hip/hip_bf16.h>
#include <stdint.h>

// ---------------------------------------------------------------------------
// GroundTruthBackward on MI455X (gfx1250).
//
// The reference collapses to: out[k(i,j)] = table[2*b + a], where
//   a = 1 iff (i,j) in edges_x_t, b = 1 iff (i,j) in edges_x_start (i<j),
//   table[2b+a] = Qt[0][1,a] * Qt[t-1][b,1] / Qt[t][b,a].
//
// Memory-bound scatter + lookup: ~33.5 MB output, ~2 MB packed codes.
// Roofline at 23.3 TB/s: ~1.6 us. No matmul structure -> WMMA inapplicable;
// we use the CDNA5 async Global->LDS path + NT streaming stores instead.
// ---------------------------------------------------------------------------

#define BLK 256

typedef unsigned int u32;
typedef u32   __attribute__((ext_vector_type(4))) u32x4;
typedef float __attribute__((ext_vector_type(4))) f32x4;

// Exact element type the async-to-LDS builtin expects (from hipcc diagnostic):
//   'int __attribute__((__vector_size__(4 * sizeof(int)))) *'
typedef int v4i __attribute__((vector_size(16)));
typedef __attribute__((address_space(1))) v4i* as1_v4i_p;
typedef __attribute__((address_space(3))) v4i* as3_v4i_p;

#if defined(__has_builtin)
#if __has_builtin(__builtin_amdgcn_global_load_async_to_lds_b128) && \
    __has_builtin(__builtin_amdgcn_s_wait_asynccnt)
#define USE_ASYNC_LDS 1
#endif
#endif
#ifndef USE_ASYNC_LDS
#define USE_ASYNC_LDS 0
#endif

__device__ __forceinline__ float sel4(u32 c, float t0, float t1, float t2, float t3) {
  float lo = (c & 1u) ? t1 : t0;   // b = 0
  float hi = (c & 1u) ? t3 : t2;   // b = 1
  return (c & 2u) ? hi : lo;
}

// Compute the 4-entry backward table from Qt (t read from device memory).
__device__ __forceinline__ void make_table(const float* __restrict__ Qt, int t,
                                           float& t0, float& t1, float& t2, float& t3) {
  const float l0 = Qt[2],               l1 = Qt[3];                // Qt[0][1, a]
  const float p0 = Qt[4 * (t - 1) + 1], p1 = Qt[4 * (t - 1) + 3]; // Qt[t-1][b, 1]
  const float q00 = Qt[4 * t + 0], q01 = Qt[4 * t + 1];
  const float q10 = Qt[4 * t + 2], q11 = Qt[4 * t + 3];           // Qt[t][b, a]
  t0 = l0 * p0 / q00;  // b=0,a=0
  t1 = l1 * p0 / q01;  // b=0,a=1
  t2 = l0 * p1 / q10;  // b=1,a=0
  t3 = l1 * p1 / q11;  // b=1,a=1
}

// --- zero fill (vectorized, handles ragged tail) ---------------------------
__global__ void zero_words_kernel(u32* __restrict__ w, u32 nwords) {
  u32 i = blockIdx.x * BLK + threadIdx.x;
  u32 base = i * 4u;
  if (base + 3u < nwords) {
    u32x4 z = {0u, 0u, 0u, 0u};
    *(u32x4*)(w + base) = z;
  } else {
    for (u32 k = base; k < nwords; ++k) w[k] = 0u;
  }
}

// --- edge scatter into 2-bit packed code array (16 codes / word) -----------
__global__ void scatter_packed_kernel(const int* __restrict__ ei, int ne,
                                      const int* __restrict__ nptr,
                                      u32* __restrict__ codes, u32 bit) {
  int e = blockIdx.x * BLK + threadIdx.x;
  if (e >= ne) return;
  int r = ei[e];
  int c = ei[ne + e];
  if (r < c) {
    int N = *nptr;
    u32 k = (u32)r * (u32)(2 * N - r - 1) / 2u + (u32)(c - r - 1);
    atomicOr(codes + (k >> 4), bit << ((k & 15u) * 2u));
  }
}

// --- edge scatter, one word per output element (fallback: codes in d_out) --
__global__ void scatter_direct_kernel(const int* __restrict__ ei, int ne,
                                      const int* __restrict__ nptr,
                                      u32* __restrict__ outw, u32 bit) {
  int e = blockIdx.x * BLK + threadIdx.x;
  if (e >= ne) return;
  int r = ei[e];
  int c = ei[ne + e];
  if (r < c) {
    int N = *nptr;
    u32 k = (u32)r * (u32)(2 * N - r - 1) / 2u + (u32)(c - r - 1);
    atomicOr(outw + k, bit);
  }
}

// --- map packed codes -> output (read ~2 MB, NT-stream ~33.5 MB) -----------
__global__ void map_packed_kernel(const float* __restrict__ Qt,
                                  const int* __restrict__ tptr,
                                  const u32* __restrict__ codes,
                                  float* __restrict__ out, u32 n) {
  const int t = *tptr;
  float t0, t1, t2, t3;
  make_table(Qt, t, t0, t1, t2, t3);

  u32 i = blockIdx.x * BLK + threadIdx.x;
  u32 base = i * 4u;
  if (base + 3u < n) {
    // base % 4 == 0, so 4 consecutive 2-bit codes live in one byte of one word.
    u32 w = codes[base >> 4];
    u32 byte = (w >> ((base & 15u) * 2u)) & 0xFFu;
    f32x4 v;
    v.x = sel4(byte & 3u,         t0, t1, t2, t3);
    v.y = sel4((byte >> 2) & 3u,  t0, t1, t2, t3);
    v.z = sel4((byte >> 4) & 3u,  t0, t1, t2, t3);
    v.w = sel4((byte >> 6) & 3u,  t0, t1, t2, t3);
    __builtin_nontemporal_store(v, (f32x4*)out + i);
  } else {
    for (u32 k = base; k < n; ++k) {
      u32 cc = (codes[k >> 4] >> ((k & 15u) * 2u)) & 3u;
      out[k] = sel4(cc, t0, t1, t2, t3);
    }
  }
}

// --- fallback: codes stored one-word-per-element in d_out, mapped in place.
//     Stages the 16B/lane code vector through LDS with the CDNA5 async path.
__global__ void map_inplace_kernel(const float* __restrict__ Qt,
                                   const int* __restrict__ tptr,
                                   float* __restrict__ out, u32 n) {
  __shared__ u32x4 tile[BLK];
  const int t = *tptr;
  float t0, t1, t2, t3;
  make_table(Qt, t, t0, t1, t2, t3);

  u32 i = blockIdx.x * BLK + threadIdx.x;
  u32 base = i * 4u;
  u32* ow = (u32*)out;
  if (base + 3u < n) {
    u32x4 code;
#if USE_ASYNC_LDS
    const u32x4* gsrc = ((const u32x4*)ow) + i;
    __builtin_amdgcn_global_load_async_to_lds_b128(
        (as1_v4i_p)(unsigned long long)(uintptr_t)gsrc,
        (as3_v4i_p)(u32)(uintptr_t)&tile[threadIdx.x],
        0, 0);
    __builtin_amdgcn_s_wait_asynccnt(0);
    code = tile[threadIdx.x];
#else
    code = ((const u32x4*)ow)[i];
#endif
    f32x4 v;
    v.x = sel4(code.x & 3u, t0, t1, t2, t3);
    v.y = sel4(code.y & 3u, t0, t1, t2, t3);
    v.z = sel4(code.z & 3u, t0, t1, t2, t3);
    v.w = sel4(code.w & 3u, t0, t1, t2, t3);
    __builtin_nontemporal_store(v, (f32x4*)out + i);
  } else {
    for (u32 k = base; k < n; ++k) {
      u32 cc = ow[k] & 3u;
      out[k] = sel4(cc, t0, t1, t2, t3);
    }
  }
}

extern "C" void kernel_launch(void* const* d_in, const int* in_sizes, int n_in,
                              void* d_out, int out_size, void* d_ws, size_t ws_size,
                              hipStream_t stream) {
  const float* Qt   = (const float*)d_in[0];
  const int*   eT   = (const int*)d_in[1];   // (2, NE) row-major
  const int*   eS   = (const int*)d_in[2];
  const int*   tptr = (const int*)d_in[3];   // scalar t
  const int*   nptr = (const int*)d_in[4];   // scalar num_nodes

  const u32 n  = (u32)out_size;              // N*(N-1)/2
  const u32 n4 = (n + 3u) / 4u;
  const int neT = in_sizes[1] / 2;
  const int neS = in_sizes[2] / 2;

  dim3 blk(BLK);
  dim3 gmap((n4 + BLK - 1) / BLK);
  dim3 geT((u32)(neT + BLK - 1) / BLK);
  dim3 geS((u32)(neS + BLK - 1) / BLK);

  const u32 ncw = (n + 15u) / 16u;           // packed code words (~2 MB)
  if (ws_size >= (size_t)ncw * sizeof(u32)) {
    // Preferred path: 2-bit packed codes in workspace (min HBM traffic).
    u32* codes = (u32*)d_ws;
    dim3 gz((((ncw + 3u) / 4u) + BLK - 1) / BLK);
    zero_words_kernel<<<gz, blk, 0, stream>>>(codes, ncw);
    scatter_packed_kernel<<<geT, blk, 0, stream>>>(eT, neT, nptr, codes, 1u); // bit a
    scatter_packed_kernel<<<geS, blk, 0, stream>>>(eS, neS, nptr, codes, 2u); // bit b
    map_packed_kernel<<<gmap, blk, 0, stream>>>(Qt, tptr, codes, (float*)d_out, n);
  } else {
    // Zero-workspace fallback: build codes in d_out, then map in place.
    u32* outw = (u32*)d_out;
    zero_words_kernel<<<gmap, blk, 0, stream>>>(outw, n);
    scatter_direct_kernel<<<geT, blk, 0, stream>>>(eT, neT, nptr, outw, 1u);
    scatter_direct_kernel<<<geS, blk, 0, stream>>>(eS, neS, nptr, outw, 2u);
    map_inplace_kernel<<<gmap, blk, 0, stream>>>(Qt, tptr, (float*)d_out, n);
  }
}